// NeighborhoodAttention2d_65171833749967
// MI455X (gfx1250) — compile-verified
//
#include <hip/hip_runtime.h>
#include <hip/hip_bf16.h>

typedef __attribute__((ext_vector_type(2))) float v2f;
typedef __attribute__((ext_vector_type(8))) float v8f;

#define CH     64
#define HEAD   16
#define IMG_H  256
#define IMG_W  512
#define HWSZ   (IMG_H * IMG_W)
#define NBATCH 8

#define TILE   16
#define HALO   18            // TILE + 2 (3x3 kernel, pad 1)
#define NHP    (HALO * HALO) // 324 halo pixels
#define NPT    21            // ceil(324/16) pixel tiles of 16

// LDS row strides (padded: conflict-free across 32 lanes, 8B-aligned pairs)
#define QROW   50            // q[0..15] k[16..31] v[32..47] per halo pixel
#define WQROW  66            // w_qkv rows (48 x 64)
#define WOROW  18            // w_out rows (64 x 16)
#define OROW   18            // attention output per interior pixel (256 x 16)

__global__ __launch_bounds__(256)
void na2d_fused_kernel(const float* __restrict__ x,
                       const float* __restrict__ wqkv,
                       const float* __restrict__ wout,
                       float* __restrict__ out) {
  __shared__ float s_qkv[NHP * QROW];     // 64.8 KB
  __shared__ float s_wq[48 * WQROW];      // 12.7 KB
  __shared__ float s_wo[CH * WOROW];      //  4.6 KB
  __shared__ float s_o[256 * OROW];       // 18.4 KB

  const int tid  = threadIdx.x;
  const int wave = tid >> 5;
  const int lane = tid & 31;
  const int n    = lane & 15;   // WMMA column (pixel) / A-row index
  const int kh   = lane >> 4;   // lane half -> K sub-offset
  const int tX = blockIdx.x, tY = blockIdx.y, bb = blockIdx.z;

  // ---- Phase 0: stage weights into LDS -----------------------------------
  for (int i = tid; i < 48 * CH; i += 256)
    s_wq[(i >> 6) * WQROW + (i & 63)] = wqkv[i];
  for (int i = tid; i < CH * HEAD; i += 256)
    s_wo[(i >> 4) * WOROW + (i & 15)] = wout[i];
  __syncthreads();

  // ---- Phase 1: QKV projection (+phi) for the 18x18 halo region ----------
  // D(48 x Npix) = w_qkv(48x64) @ x(64 x Npix), K=64 as 16 chained f32 WMMAs
  for (int t = wave; t < NPT; t += 8) {
    const int p  = t * 16 + n;                  // halo pixel id for this lane
    const int pc = p < NHP ? p : NHP - 1;
    const int hh = pc / HALO, ww = pc - hh * HALO;
    const int gh = tY * TILE + hh - 1;
    const int gw = tX * TILE + ww - 1;
    const bool inimg = (p < NHP) && (gh >= 0) && (gh < IMG_H) &&
                       (gw >= 0) && (gw < IMG_W);
    const int ghc = gh < 0 ? 0 : (gh >= IMG_H ? IMG_H - 1 : gh);
    const int gwc = gw < 0 ? 0 : (gw >= IMG_W ? IMG_W - 1 : gw);
    const float* xp = x + ((size_t)bb * CH) * HWSZ + (size_t)ghc * IMG_W + gwc;

    // Prefetch B-fragments for all 16 K-steps (clamped addr + select->0)
    float bf[32];
#pragma unroll
    for (int s = 0; s < 16; ++s) {
      const int c0 = 4 * s + 2 * kh;
      const float a0 = xp[(size_t)c0 * HWSZ];
      const float a1 = xp[(size_t)(c0 + 1) * HWSZ];
      bf[2 * s]     = inimg ? a0 : 0.f;
      bf[2 * s + 1] = inimg ? a1 : 0.f;
    }

    v8f acc0 = {0.f,0.f,0.f,0.f,0.f,0.f,0.f,0.f};
    v8f acc1 = {0.f,0.f,0.f,0.f,0.f,0.f,0.f,0.f};
    v8f acc2 = {0.f,0.f,0.f,0.f,0.f,0.f,0.f,0.f};
#pragma unroll
    for (int s = 0; s < 16; ++s) {
      const int kb = 4 * s + 2 * kh;
      v2f bv = { bf[2 * s], bf[2 * s + 1] };
      v2f a0 = { s_wq[( 0 + n) * WQROW + kb], s_wq[( 0 + n) * WQROW + kb + 1] };
      v2f a1 = { s_wq[(16 + n) * WQROW + kb], s_wq[(16 + n) * WQROW + kb + 1] };
      v2f a2 = { s_wq[(32 + n) * WQROW + kb], s_wq[(32 + n) * WQROW + kb + 1] };
      acc0 = __builtin_amdgcn_wmma_f32_16x16x4_f32(false, a0, false, bv,
                                                   (short)0, acc0, false, false);
      acc1 = __builtin_amdgcn_wmma_f32_16x16x4_f32(false, a1, false, bv,
                                                   (short)0, acc1, false, false);
      acc2 = __builtin_amdgcn_wmma_f32_16x16x4_f32(false, a2, false, bv,
                                                   (short)0, acc2, false, false);
    }

    // phi(t) = elu(t)+1 = t+1 (t>0) else exp(t); zero q/k/v outside image
    const bool st = p < NHP;
#pragma unroll
    for (int r = 0; r < 8; ++r) {
      const int row = r + 8 * kh;                // D-matrix M index
      float qv = acc0[r]; qv = qv > 0.f ? qv + 1.f : __expf(qv);
      float kv = acc1[r]; kv = kv > 0.f ? kv + 1.f : __expf(kv);
      float vv = acc2[r];
      if (!inimg) { qv = 0.f; kv = 0.f; vv = 0.f; }
      if (st) {
        s_qkv[p * QROW +      row] = qv;
        s_qkv[p * QROW + 16 + row] = kv;
        s_qkv[p * QROW + 32 + row] = vv;
      }
    }
  }
  __syncthreads();

  // ---- Phase 2: per-pixel 3x3 linear attention (one thread per pixel) ----
  {
    const int pr = tid >> 4, pcid = tid & 15;
    const int p = (pr + 1) * HALO + (pcid + 1);   // interior -> halo coords
    float q[HEAD];
#pragma unroll
    for (int d = 0; d < HEAD; ++d) q[d] = s_qkv[p * QROW + d];
    float sv[HEAD];
#pragma unroll
    for (int d = 0; d < HEAD; ++d) sv[d] = 0.f;
    float ss = 0.f;
#pragma unroll
    for (int dy = -1; dy <= 1; ++dy) {
#pragma unroll
      for (int dx = -1; dx <= 1; ++dx) {
        const int np = p + dy * HALO + dx;
        const float* kp = &s_qkv[np * QROW + 16];
        float sc = 0.f;
#pragma unroll
        for (int d = 0; d < HEAD; ++d) sc += q[d] * kp[d];
        ss += sc;
        const float* vp = kp + 16;
#pragma unroll
        for (int d = 0; d < HEAD; ++d) sv[d] += sc * vp[d];
      }
    }
    const float inv = 1.f / (ss + 1e-6f);
#pragma unroll
    for (int d = 0; d < HEAD; ++d) s_o[tid * OROW + d] = sv[d] * inv;
  }
  __syncthreads();

  // ---- Phase 3: output projection, D(64 x 256px) = w_out(64x16) @ o ------
  for (int t = wave; t < 16; t += 8) {
    const int p = t * 16 + n;          // interior pixel (row t, col n of tile)
    const int kb0 = 2 * kh;
    v2f b0 = { s_o[p * OROW +      kb0], s_o[p * OROW +      kb0 + 1] };
    v2f b1 = { s_o[p * OROW +  4 + kb0], s_o[p * OROW +  4 + kb0 + 1] };
    v2f b2 = { s_o[p * OROW +  8 + kb0], s_o[p * OROW +  8 + kb0 + 1] };
    v2f b3 = { s_o[p * OROW + 12 + kb0], s_o[p * OROW + 12 + kb0 + 1] };
    const int gRow = tY * TILE + t;
    const int gCol = tX * TILE + n;
    float* op = out + ((size_t)bb * CH) * HWSZ + (size_t)gRow * IMG_W + gCol;
#pragma unroll
    for (int m = 0; m < 4; ++m) {
      const float* wr = &s_wo[(m * 16 + n) * WOROW];
      v8f acc = {0.f,0.f,0.f,0.f,0.f,0.f,0.f,0.f};
      v2f a0 = { wr[     kb0], wr[     kb0 + 1] };
      acc = __builtin_amdgcn_wmma_f32_16x16x4_f32(false, a0, false, b0,
                                                  (short)0, acc, false, false);
      v2f a1 = { wr[ 4 + kb0], wr[ 4 + kb0 + 1] };
      acc = __builtin_amdgcn_wmma_f32_16x16x4_f32(false, a1, false, b1,
                                                  (short)0, acc, false, false);
      v2f a2 = { wr[ 8 + kb0], wr[ 8 + kb0 + 1] };
      acc = __builtin_amdgcn_wmma_f32_16x16x4_f32(false, a2, false, b2,
                                                  (short)0, acc, false, false);
      v2f a3 = { wr[12 + kb0], wr[12 + kb0 + 1] };
      acc = __builtin_amdgcn_wmma_f32_16x16x4_f32(false, a3, false, b3,
                                                  (short)0, acc, false, false);
#pragma unroll
      for (int r = 0; r < 8; ++r) {
        const int O = m * 16 + r + 8 * kh;       // output channel
        op[(size_t)O * HWSZ] = acc[r];
      }
    }
  }
}

extern "C" void kernel_launch(void* const* d_in, const int* in_sizes, int n_in,
                              void* d_out, int out_size, void* d_ws, size_t ws_size,
                              hipStream_t stream) {
  (void)in_sizes; (void)n_in; (void)out_size; (void)d_ws; (void)ws_size;
  const float* x     = (const float*)d_in[0];
  const float* wqkv  = (const float*)d_in[1];
  const float* wout  = (const float*)d_in[2];
  float* out         = (float*)d_out;

  dim3 grid(IMG_W / TILE, IMG_H / TILE, NBATCH);   // 32 x 16 x 8 = 4096 WGs
  dim3 block(256);                                  // 8 wave32
  na2d_fused_kernel<<<grid, block, 0, stream>>>(x, wqkv, wout, out);
}